// GraphSAGE_32976758899003
// MI455X (gfx1250) — compile-verified
//
#include <hip/hip_runtime.h>
#include <hip/hip_bf16.h>

typedef __attribute__((ext_vector_type(2))) float v2f;
typedef __attribute__((ext_vector_type(4))) float v4f;
typedef __attribute__((ext_vector_type(8))) float v8f;

#define N_NODES 100000
#define F_IN    256
#define HDIM    128
#define OUT_F   64
#define LDA     260   // LDS row stride (floats) for K=256 tiles: 260 mod 64 == 4 -> conflict-free b64 reads
#define LDH     132   // LDS row stride (floats) for K=128 tiles
#define LN_EPS  1e-5f

// ---------------------------------------------------------------------------
// One 16x16 fp32 WMMA tile over K (K multiple of 4), A from LDS, B from global.
// A frag (16x4 f32): lane<16 -> M=lane, K=k0,k0+1 ; lane>=16 -> M=lane-16, K=k0+2,k0+3
// B frag (4x16 f32): lane<16 -> N=lane, K=k0,k0+1 ; lane>=16 -> N=lane-16, K=k0+2,k0+3
// ---------------------------------------------------------------------------
__device__ __forceinline__ v8f wmma_tile_f32(const float* __restrict__ Alds, int lda,
                                             const float* __restrict__ W, int ldw,
                                             int colBase, int K, int lane) {
    const int m    = lane & 15;
    const int n    = lane & 15;
    const int ksel = (lane >> 4) * 2;     // 0 or 2
    v8f c = {};
    for (int k0 = 0; k0 < K; k0 += 4) {
        v2f a = *(const v2f*)(Alds + m * lda + k0 + ksel);
        v2f b;
        b.x = W[(k0 + ksel)     * ldw + colBase + n];
        b.y = W[(k0 + ksel + 1) * ldw + colBase + n];
        c = __builtin_amdgcn_wmma_f32_16x16x4_f32(false, a, false, b, (short)0, c,
                                                  false, false);
    }
    return c;
}

// ---------------------------------------------------------------------------
// Kernel 1: h = X @ W1 + b1    [N,256]x[256,128] -> [N,128]
// 256 threads = 8 waves; 16 rows per block; wave w computes column tile w*16.
// ---------------------------------------------------------------------------
__global__ __launch_bounds__(256) void k_gemm1(const float* __restrict__ X,
                                               const float* __restrict__ W1,
                                               const float* __restrict__ b1,
                                               float* __restrict__ hout) {
    __shared__ float As[16 * LDA];
    const int rowBase = blockIdx.x * 16;
    const int tid = threadIdx.x;

    // stage 16x256 A tile (1024 float4 slots, 64 per row)
    for (int i = tid; i < 16 * (F_IN / 4); i += 256) {
        const int r = i >> 6, c4 = i & 63;
        *(v4f*)(As + r * LDA + c4 * 4) =
            *(const v4f*)(X + (size_t)(rowBase + r) * F_IN + c4 * 4);
    }
    __syncthreads();

    const int wave = tid >> 5, lane = tid & 31;
    const int colBase = wave * 16;
    v8f c = wmma_tile_f32(As, LDA, W1, HDIM, colBase, F_IN, lane);

    const int n = lane & 15;
    const int rOff = (lane >> 4) * 8;
    for (int r = 0; r < 8; ++r) {
        const int col = colBase + n;
        hout[(size_t)(rowBase + r + rOff) * HDIM + col] = c[r] + b1[col];
    }
}

// ---------------------------------------------------------------------------
// Kernel 2: one SAGE layer.
//   agg[i] = sum_{t<k} hin[sn[i*k+t]]           (random 512B row gathers)
//   h      = LN(relu([hin_i | agg_i] @ W2 + b2))
// 256 threads = 8 waves, 16 rows/block. Indices staged in LDS (no 32x
// redundant idx loads); gathers are coalesced b128 with prefetch look-ahead.
// ---------------------------------------------------------------------------
__global__ __launch_bounds__(256) void k_sage_layer(const float* __restrict__ hin,
                                                    const int* __restrict__ sn,
                                                    const float* __restrict__ W2,
                                                    const float* __restrict__ b2,
                                                    const float* __restrict__ gamma,
                                                    const float* __restrict__ beta,
                                                    float* __restrict__ hout,
                                                    int K_FAN) {
    __shared__ float As[16 * LDA];     // [own | agg]  16 x 256
    __shared__ float Act[16 * LDH];    // post-relu pre-LN 16 x 128
    __shared__ int   Idx[16 * 50];     // neighbor indices (max fanout 50)
    __shared__ float Mu[16], Rs[16];

    const int rowBase = blockIdx.x * 16;
    const int tid = threadIdx.x;

    // stage neighbor indices for the whole block
    const int totIdx = 16 * K_FAN;
    for (int i = tid; i < totIdx; i += 256)
        Idx[i] = sn[(size_t)rowBase * K_FAN + i];
    __syncthreads();

    const int wave = tid >> 5, lane = tid & 31;

    // gather + own-row staging: each wave owns 2 rows; lane covers 4 floats
    for (int rr = 0; rr < 2; ++rr) {
        const int r = wave * 2 + rr;
        *(v4f*)(As + r * LDA + lane * 4) =
            *(const v4f*)(hin + (size_t)(rowBase + r) * HDIM + lane * 4);
        v4f acc = {};
        for (int t = 0; t < K_FAN; ++t) {
            const int idx = Idx[r * K_FAN + t];
            if (t + 2 < K_FAN)  // hide random-access latency
                __builtin_prefetch(hin + (size_t)Idx[r * K_FAN + t + 2] * HDIM + lane * 4, 0, 0);
            acc += *(const v4f*)(hin + (size_t)idx * HDIM + lane * 4);
        }
        *(v4f*)(As + r * LDA + HDIM + lane * 4) = acc;
    }
    __syncthreads();

    // concat-GEMM: [16 x 256] @ [256 x 128]
    const int colBase = wave * 16;
    v8f c = wmma_tile_f32(As, LDA, W2, HDIM, colBase, 2 * HDIM, lane);

    // bias + relu -> LDS
    const int n = lane & 15;
    const int rOff = (lane >> 4) * 8;
    for (int r = 0; r < 8; ++r) {
        float v = c[r] + b2[colBase + n];
        Act[(r + rOff) * LDH + colBase + n] = v > 0.f ? v : 0.f;
    }
    __syncthreads();

    // LayerNorm stats: one thread per row (2048 LDS reads total, trivial)
    if (tid < 16) {
        float s = 0.f, s2 = 0.f;
        for (int j = 0; j < HDIM; ++j) {
            const float v = Act[tid * LDH + j];
            s += v; s2 += v * v;
        }
        const float mu  = s * (1.f / HDIM);
        const float var = s2 * (1.f / HDIM) - mu * mu;
        Mu[tid] = mu;
        Rs[tid] = rsqrtf(var + LN_EPS);
    }
    __syncthreads();

    // apply LN + affine, write out
    for (int i = tid; i < 16 * HDIM; i += 256) {
        const int r = i >> 7, j = i & (HDIM - 1);
        hout[(size_t)(rowBase + r) * HDIM + j] =
            (Act[r * LDH + j] - Mu[r]) * Rs[r] * gamma[j] + beta[j];
    }
}

// ---------------------------------------------------------------------------
// Kernel 3: out = h @ W3 + b3   [N,128]x[128,64] -> [N,64]
// 128 threads = 4 waves (one per 16-col tile), 16 rows per block.
// ---------------------------------------------------------------------------
__global__ __launch_bounds__(128) void k_gemm3(const float* __restrict__ hin,
                                               const float* __restrict__ W3,
                                               const float* __restrict__ b3,
                                               float* __restrict__ out) {
    __shared__ float As[16 * LDH];
    const int rowBase = blockIdx.x * 16;
    const int tid = threadIdx.x;

    for (int i = tid; i < 16 * (HDIM / 4); i += 128) {
        const int r = i >> 5, c4 = i & 31;
        *(v4f*)(As + r * LDH + c4 * 4) =
            *(const v4f*)(hin + (size_t)(rowBase + r) * HDIM + c4 * 4);
    }
    __syncthreads();

    const int wave = tid >> 5, lane = tid & 31;
    const int colBase = wave * 16;
    v8f c = wmma_tile_f32(As, LDH, W3, OUT_F, colBase, HDIM, lane);

    const int n = lane & 15;
    const int rOff = (lane >> 4) * 8;
    for (int r = 0; r < 8; ++r) {
        const int col = colBase + n;
        out[(size_t)(rowBase + r + rOff) * OUT_F + col] = c[r] + b3[col];
    }
}

// ---------------------------------------------------------------------------
extern "C" void kernel_launch(void* const* d_in, const int* in_sizes, int n_in,
                              void* d_out, int out_size, void* d_ws, size_t ws_size,
                              hipStream_t stream) {
    const float* X     = (const float*)d_in[0];
    // d_in[1] = edge_index (unused by the reference computation)
    const int*   sn0   = (const int*)d_in[2];
    const int*   sn1   = (const int*)d_in[3];
    const float* W1    = (const float*)d_in[4];
    const float* b1    = (const float*)d_in[5];
    const float* W2    = (const float*)d_in[6];
    const float* b2    = (const float*)d_in[7];
    const float* gamma = (const float*)d_in[8];
    const float* beta  = (const float*)d_in[9];
    const float* W3    = (const float*)d_in[10];
    const float* b3    = (const float*)d_in[11];
    float* out  = (float*)d_out;

    float* bufA = (float*)d_ws;                       // N x 128
    float* bufB = bufA + (size_t)N_NODES * HDIM;      // N x 128  (ping-pong)

    const int nblk = N_NODES / 16;                    // 6250, exact

    k_gemm1<<<nblk, 256, 0, stream>>>(X, W1, b1, bufA);
    k_sage_layer<<<nblk, 256, 0, stream>>>(bufA, sn0, W2, b2, gamma, beta, bufB, 10);
    k_sage_layer<<<nblk, 256, 0, stream>>>(bufB, sn1, W2, b2, gamma, beta, bufA, 50);
    k_gemm3<<<nblk, 128, 0, stream>>>(bufA, W3, b3, out);
}